// Global_Gen_Sum_Mean_Max_Pool_53979148976325
// MI455X (gfx1250) — compile-verified
//
#include <hip/hip_runtime.h>
#include <stdint.h>
#include <math.h>

#define N_FEAT  128
#define TR      4     // rows per tile (one row per wave32)
#define THREADS 128   // 4 waves
#define NBUF    4     // pipeline depth (async tiles in flight)

typedef int v4i __attribute__((ext_vector_type(4)));
typedef __attribute__((address_space(1))) v4i* gv4i_p;  // global int4*
typedef __attribute__((address_space(3))) v4i* lv4i_p;  // LDS int4*

#if defined(__has_builtin)
#  if __has_builtin(__builtin_amdgcn_global_load_async_to_lds_b128)
#    define HAVE_ASYNC_BUILTIN 1
#  endif
#  if __has_builtin(__builtin_amdgcn_s_wait_asynccnt)
#    define HAVE_WAIT_BUILTIN 1
#  endif
#endif
#ifndef HAVE_ASYNC_BUILTIN
#  define HAVE_ASYNC_BUILTIN 0
#endif
#ifndef HAVE_WAIT_BUILTIN
#  define HAVE_WAIT_BUILTIN 0
#endif

#if HAVE_WAIT_BUILTIN
#  define WAIT_ASYNC(n) __builtin_amdgcn_s_wait_asynccnt(n)
#else
#  define WAIT_ASYNC(n) asm volatile("s_wait_asynccnt " #n ::: "memory")
#endif

// ---------------------------------------------------------------------------
// Kernel 1: segment starts via per-graph binary search on sorted int64 batch.
// starts[g] = lower_bound(batch, g); starts[B] = N.
// ---------------------------------------------------------------------------
__global__ void seg_starts_k(const long long* __restrict__ batch, int n, int bsize,
                             int* __restrict__ starts) {
  int g = blockIdx.x * blockDim.x + threadIdx.x;
  if (g > bsize) return;
  long long target = (long long)g;
  int lo = 0, hi = n;
  while (lo < hi) {
    int mid = (lo + hi) >> 1;
    if (batch[mid] < target) lo = mid + 1; else hi = mid;
  }
  starts[g] = lo;
}

// ---------------------------------------------------------------------------
// Kernel 2: one block per graph. Single-pass online softmax pooling.
// Wave w handles row (tile_base + w); lane owns 4 features. Tiles are staged
// global -> LDS with async b128 copies, NBUF-deep pipeline on ASYNCcnt.
// ---------------------------------------------------------------------------
__global__ __launch_bounds__(THREADS)
void softmax_pool_k(const float* __restrict__ x, const int* __restrict__ starts,
                    const float* __restrict__ p_ptr, const float* __restrict__ beta_ptr,
                    float* __restrict__ out) {
  __shared__ __align__(16) float buf[NBUF][TR * N_FEAT];  // 4 x 2KB tiles
  __shared__ float rm[THREADS][4];
  __shared__ float rd[THREADS][4];
  __shared__ float rs[THREADS][4];

  const int g     = blockIdx.x;
  const int t     = threadIdx.x;
  const int start = starts[g];
  const int end   = starts[g + 1];
  const int n     = end - start;
  const float p    = p_ptr[0];
  const float beta = beta_ptr[0];
  const int grp = t >> 5;          // wave index -> row offset in tile
  const int col = (t & 31) << 2;   // 4 features per lane

  // online softmax state, one per owned feature
  float m0 = -INFINITY, m1 = -INFINITY, m2 = -INFINITY, m3 = -INFINITY;
  float d0 = 0.f, d1 = 0.f, d2 = 0.f, d3 = 0.f;
  float s0 = 0.f, s1 = 0.f, s2 = 0.f, s3 = 0.f;

#define UPD(mm, dd, ss, vv)                         \
  { float l  = p * (vv);                            \
    float nm = fmaxf(mm, l);                        \
    float a  = __expf(mm - nm);                     \
    float b  = __expf(l - nm);                      \
    dd = dd * a + b;                                \
    ss = ss * a + (vv) * b;                         \
    mm = nm; }

  if (n > 0) {
    const int iters = (n + TR - 1) / TR;

    auto issue = [&](int it, int b) {
      int row = start + it * TR + grp;
      if (row > end - 1) row = end - 1;              // clamp tail (data ignored)
      const float* gp = x + (size_t)row * N_FEAT + col;
      float* lp = &buf[b][t << 2];
#if HAVE_ASYNC_BUILTIN
      __builtin_amdgcn_global_load_async_to_lds_b128(
          (gv4i_p)(unsigned long long)(uintptr_t)gp,   // flat global == AS1 addr
          (lv4i_p)(unsigned)(uintptr_t)lp,             // low 32 bits == LDS offset
          0, 0);
#else
      unsigned lds_off = (unsigned)(uintptr_t)lp;
      unsigned long long gaddr = (unsigned long long)(uintptr_t)gp;
      asm volatile("global_load_async_to_lds_b128 %0, %1, off"
                   :: "v"(lds_off), "v"(gaddr) : "memory");
#endif
    };

    // prologue: fill pipeline (tiles 0..NBUF-2)
    issue(0, 0);
    if (1 < iters) issue(1, 1);
    if (2 < iters) issue(2, 2);

    for (int it = 0; it < iters; ++it) {
      const int b  = it & (NBUF - 1);
      const int nx = it + (NBUF - 1);                // tile issued this iteration
      if (nx < iters) {
        issue(nx, nx & (NBUF - 1));
        WAIT_ASYNC(3);                               // tiles <= it complete (in-order)
      } else {
        const int rem = iters - 1 - it;              // tiles still outstanding past it
        if      (rem >= 2) WAIT_ASYNC(2);
        else if (rem == 1) WAIT_ASYNC(1);
        else               WAIT_ASYNC(0);
      }
      const int row = start + it * TR + grp;
      if (row < end) {
        float4 v = *(const float4*)&buf[b][t << 2];  // lane reads only its own 16B
        UPD(m0, d0, s0, v.x);
        UPD(m1, d1, s1, v.y);
        UPD(m2, d2, s2, v.z);
        UPD(m3, d3, s3, v.w);
      }
    }
  }
#undef UPD

  rm[t][0] = m0; rm[t][1] = m1; rm[t][2] = m2; rm[t][3] = m3;
  rd[t][0] = d0; rd[t][1] = d1; rd[t][2] = d2; rd[t][3] = d3;
  rs[t][0] = s0; rs[t][1] = s1; rs[t][2] = s2; rs[t][3] = s3;
  __syncthreads();

  if (t < 32) {
    const float nf    = (float)n;
    const float scale = nf / (1.0f + beta * (nf - 1.0f));
    float res[4];
#pragma unroll
    for (int k = 0; k < 4; ++k) {
      float M = rm[t][k];
      M = fmaxf(M, rm[t + 32][k]);
      M = fmaxf(M, rm[t + 64][k]);
      M = fmaxf(M, rm[t + 96][k]);
      float D = 0.f, S = 0.f;
#pragma unroll
      for (int j = 0; j < 4; ++j) {
        float a = __expf(rm[t + 32 * j][k] - M);   // groups with no rows: a==0
        D += rd[t + 32 * j][k] * a;
        S += rs[t + 32 * j][k] * a;
      }
      res[k] = (n > 0) ? (S / D) * scale : 0.0f * scale;  // n==0, beta==1 -> NaN (matches ref)
    }
    float4 o = make_float4(res[0], res[1], res[2], res[3]);
    *(float4*)(out + (size_t)g * N_FEAT + (t << 2)) = o;
  }
}

// ---------------------------------------------------------------------------
extern "C" void kernel_launch(void* const* d_in, const int* in_sizes, int n_in,
                              void* d_out, int out_size, void* d_ws, size_t ws_size,
                              hipStream_t stream) {
  (void)n_in; (void)ws_size;
  const float*     x     = (const float*)d_in[0];
  const long long* batch = (const long long*)d_in[1];  // int64 per reference
  const float*     p     = (const float*)d_in[2];
  const float*     beta  = (const float*)d_in[3];
  // d_in[4] (bsize) lives on device; derive B from out_size instead.
  const int N = in_sizes[1];
  const int B = out_size / N_FEAT;

  int* starts = (int*)d_ws;                            // (B+1) ints of scratch

  const int st_threads = 256;
  const int st_blocks  = (B + 1 + st_threads - 1) / st_threads;
  seg_starts_k<<<st_blocks, st_threads, 0, stream>>>(batch, N, B, starts);
  softmax_pool_k<<<B, THREADS, 0, stream>>>(x, starts, p, beta, (float*)d_out);
}